// GraphResnetBlock_13795434955523
// MI455X (gfx1250) — compile-verified
//
#include <hip/hip_runtime.h>
#include <cstddef>

#define N_NODES 50000
#define N_EDGES 600000
#define DIM_IN  128
#define DIM_OUT 256
#define EPS_BN  1e-5f

typedef __attribute__((ext_vector_type(2))) float v2f;
typedef __attribute__((ext_vector_type(8))) float v8f;

// ---------------------------------------------------------------------------
// Kernel 0: zero agg + column accumulators (float4 grid-stride)
// ---------------------------------------------------------------------------
__global__ void k_zero(float* __restrict__ p, int n4) {
  float4 z = make_float4(0.f, 0.f, 0.f, 0.f);
  for (int i = blockIdx.x * blockDim.x + threadIdx.x; i < n4;
       i += gridDim.x * blockDim.x) {
    reinterpret_cast<float4*>(p)[i] = z;
  }
}

// ---------------------------------------------------------------------------
// Kernel 1: COO scatter-sum. One wave32 per edge; each lane moves float4
// (32 lanes * 16B = 512B = one 128-float feature row) via f32 global atomics.
// x rows (25.6 MB) are L2-resident on MI455X (192 MB L2).
// ---------------------------------------------------------------------------
__global__ void __launch_bounds__(256) k_scatter(
    const float* __restrict__ x, const int* __restrict__ edges,
    float* __restrict__ agg) {
  const int lane = threadIdx.x & 31;
  const int wib  = threadIdx.x >> 5;
  const int e    = blockIdx.x * 8 + wib;
  if (e >= N_EDGES) return;
  const int s = edges[e];
  const int d = edges[N_EDGES + e];
  float4 v = reinterpret_cast<const float4*>(x + (size_t)s * DIM_IN)[lane];
  float* dp = agg + (size_t)d * DIM_IN + lane * 4;
  unsafeAtomicAdd(dp + 0, v.x);
  unsafeAtomicAdd(dp + 1, v.y);
  unsafeAtomicAdd(dp + 2, v.z);
  unsafeAtomicAdd(dp + 3, v.w);
}

// ---------------------------------------------------------------------------
// Shared WMMA 16x16 (M,N) x K=128 fp32 tile: 32x V_WMMA_F32_16X16X4_F32.
// A lane L: row = L%16, k-pair = (L/16)*2 (ISA 7.12.2, 32-bit A 16x4).
// B lane L: col = L%16, k-pair = (L/16)*2 (B row striped across lanes).
// C/D: VGPR r -> row r + 8*(L/16), col L%16.
// ---------------------------------------------------------------------------
__device__ inline v8f wmma_tile_f32(const float* __restrict__ A,   // [.,128] row-major
                                    const float* __restrict__ B,   // [128,256] row-major
                                    int m0, int n0, int lane) {
  const int lm = lane & 15;
  const int lh = lane >> 4;
  const float* arow = A + (size_t)(m0 + lm) * DIM_IN + lh * 2;
  const float* bcol = B + (size_t)(lh * 2) * DIM_OUT + n0 + lm;
  v8f acc = {};
#pragma unroll
  for (int kk = 0; kk < DIM_IN; kk += 4) {
    v2f a;
    a.x = arow[kk];
    a.y = arow[kk + 1];
    v2f b;
    b.x = bcol[(size_t)kk * DIM_OUT];
    b.y = bcol[(size_t)(kk + 1) * DIM_OUT];
    acc = __builtin_amdgcn_wmma_f32_16x16x4_f32(false, a, false, b,
                                                (short)0, acc, false, false);
  }
  return acc;
}

// ---------------------------------------------------------------------------
// Kernel 2: out_pre = agg @ W_gcn + b_gcn  -> d_out (temp storage),
// plus per-column sum / sum-of-squares for batchnorm stats.
// Grid: (3125, 2) x 256 threads; wave w handles n-tile blockIdx.y*8+w.
// ---------------------------------------------------------------------------
__global__ void __launch_bounds__(256) k_gemm_gcn(
    const float* __restrict__ agg, const float* __restrict__ W,
    const float* __restrict__ bias, float* __restrict__ out_pre,
    float* __restrict__ colsum, float* __restrict__ colsumsq) {
  const int lane = threadIdx.x & 31;
  const int wave = threadIdx.x >> 5;
  const int m0 = blockIdx.x * 16;
  const int n0 = (blockIdx.y * 8 + wave) * 16;
  const int lm = lane & 15;
  const int lh = lane >> 4;

  v8f acc = wmma_tile_f32(agg, W, m0, n0, lane);

  const int col = n0 + lm;
  const float bia = bias[col];
  float s = 0.f, sq = 0.f;
#pragma unroll
  for (int r = 0; r < 8; ++r) {
    float v = acc[r] + bia;
    const int row = m0 + r + lh * 8;
    out_pre[(size_t)row * DIM_OUT + col] = v;
    s += v;
    sq += v * v;
  }
  // lanes L and L+16 hold the same column -> fold, then 16 atomics per wave
  s  += __shfl_xor(s, 16);
  sq += __shfl_xor(sq, 16);
  if (lane < 16) {
    unsafeAtomicAdd(&colsum[col], s);
    unsafeAtomicAdd(&colsumsq[col], sq);
  }
}

// ---------------------------------------------------------------------------
// Kernel 3: fold stats -> per-channel scale/shift  (256 threads)
// ---------------------------------------------------------------------------
__global__ void k_stats(const float* __restrict__ colsum,
                        const float* __restrict__ colsumsq,
                        const float* __restrict__ gamma,
                        const float* __restrict__ beta,
                        float* __restrict__ scale, float* __restrict__ shift) {
  const int n = threadIdx.x;
  const float inv_n = 1.0f / (float)N_NODES;
  float mean = colsum[n] * inv_n;
  float var  = colsumsq[n] * inv_n - mean * mean;
  float sc   = gamma[n] * rsqrtf(var + EPS_BN);
  scale[n] = sc;
  shift[n] = beta[n] - mean * sc;
}

// ---------------------------------------------------------------------------
// Kernel 4: sc = x @ W_sc + b_sc; out = sc + elu(bn(out_pre)), in-place on
// d_out (each element read+written by exactly one lane).
// ---------------------------------------------------------------------------
__global__ void __launch_bounds__(256) k_gemm_sc(
    const float* __restrict__ x, const float* __restrict__ W,
    const float* __restrict__ bias, const float* __restrict__ scale,
    const float* __restrict__ shift, float* __restrict__ out) {
  const int lane = threadIdx.x & 31;
  const int wave = threadIdx.x >> 5;
  const int m0 = blockIdx.x * 16;
  const int n0 = (blockIdx.y * 8 + wave) * 16;
  const int lm = lane & 15;
  const int lh = lane >> 4;

  v8f acc = wmma_tile_f32(x, W, m0, n0, lane);

  const int col = n0 + lm;
  const float bia = bias[col];
  const float scl = scale[col];
  const float shf = shift[col];
#pragma unroll
  for (int r = 0; r < 8; ++r) {
    const int row = m0 + r + lh * 8;
    const size_t idx = (size_t)row * DIM_OUT + col;
    float h = out[idx] * scl + shf;           // batchnorm affine
    float act = h > 0.f ? h : expm1f(h);      // ELU (alpha=1)
    out[idx] = acc[r] + bia + act;            // shortcut + residual
  }
}

// ---------------------------------------------------------------------------
extern "C" void kernel_launch(void* const* d_in, const int* in_sizes, int n_in,
                              void* d_out, int out_size, void* d_ws,
                              size_t ws_size, hipStream_t stream) {
  (void)in_sizes; (void)n_in; (void)out_size; (void)ws_size;
  const float* x     = (const float*)d_in[0];
  const int*   edges = (const int*)d_in[1];
  const float* W_gcn = (const float*)d_in[2];
  const float* b_gcn = (const float*)d_in[3];
  const float* gamma = (const float*)d_in[4];
  const float* beta  = (const float*)d_in[5];
  const float* W_sc  = (const float*)d_in[6];
  const float* b_sc  = (const float*)d_in[7];
  float* out = (float*)d_out;

  // ws layout: agg[50000*128] | colsum[256] | colsumsq[256] | scale[256] | shift[256]
  float* agg      = (float*)d_ws;
  float* colsum   = agg + (size_t)N_NODES * DIM_IN;
  float* colsumsq = colsum + DIM_OUT;
  float* scale    = colsumsq + DIM_OUT;
  float* shift    = scale + DIM_OUT;

  const int zero_n4 = (N_NODES * DIM_IN + 2 * DIM_OUT) / 4;  // 1,600,128 float4s
  k_zero<<<2048, 256, 0, stream>>>(agg, zero_n4);

  k_scatter<<<N_EDGES / 8, 256, 0, stream>>>(x, edges, agg);

  dim3 gemm_grid(N_NODES / 16, 2);  // 3125 x 2, 8 waves/block -> 16 n-tiles
  k_gemm_gcn<<<gemm_grid, 256, 0, stream>>>(agg, W_gcn, b_gcn, out, colsum,
                                            colsumsq);

  k_stats<<<1, DIM_OUT, 0, stream>>>(colsum, colsumsq, gamma, beta, scale,
                                     shift);

  k_gemm_sc<<<gemm_grid, 256, 0, stream>>>(x, W_sc, b_sc, scale, shift, out);
}